// Linear_82824149336078
// MI455X (gfx1250) — compile-verified
//
#include <hip/hip_runtime.h>

typedef __attribute__((ext_vector_type(2))) float v2f;
typedef __attribute__((ext_vector_type(8))) float v8f;

#define FEAT 2304      // total feature dim (256*1 + 256*3 + 256*5)
#define NCOL 256       // N (output multiplicity) per block
#define KDIM 256       // K (input multiplicity) per block
#define KC   64        // K-chunk staged through LDS
#define NCHUNK (KDIM / KC)   // 4
#define ZT   16        // batch rows per workgroup
#define BATCH 50000

// CDNA5 async global->LDS copy (ASYNCcnt-tracked, no VGPR staging).
// LDS destination address is taken from the vdst VGPR; low 32 bits of a flat LDS pointer == LDS offset.
#define ASYNC_CP_B128(lds_off, gptr) \
    asm volatile("global_load_async_to_lds_b128 %0, %1, off" \
                 :: "v"(lds_off), "v"(gptr) : "memory")
#define WAIT_ASYNC(n) \
    asm volatile("s_wait_asynccnt %0" :: "n"(n) : "memory")

// One irrep block: Y[z, w*D+i] = (1/16) * sum_u W[u,w] * X[z, u*D+i]
// GEMM view: rows r = z*D+i (M = ZT*D per WG), cols w (N=256), K=u (256).
template <int D>
__global__ __launch_bounds__(256)
void linear_block_kernel(const float* __restrict__ x,
                         const float* __restrict__ wgt,
                         float* __restrict__ y,
                         int xoff, int woff)
{
    // Double-buffered LDS stage: per buffer 16 z-rows x KC u's x D i's
    __shared__ __align__(16) float lx[2][ZT * KC * D];

    const int tid  = threadIdx.x;
    const int lane = tid & 31;
    const int wave = tid >> 5;      // 0..7 -> N strip
    const int hi   = lane >> 4;     // lane half (K split in A/B fragments)
    const int l16  = lane & 15;

    const int z0 = blockIdx.x * ZT;
    const float* __restrict__ xblk = x + (size_t)z0 * FEAT + xoff;
    const float* __restrict__ wblk = wgt + woff;

    // Per-thread staging slots: idx = tid + 256*j covers 256*D float4s/chunk
    const float* ssrc[D];
    unsigned     sdst[D];
#pragma unroll
    for (int j = 0; j < D; ++j) {
        int idx = tid + 256 * j;
        int z   = idx / (16 * D);         // 16*D float4s per z-row
        int q   = idx % (16 * D);
        ssrc[j] = xblk + (size_t)z * FEAT + 4 * q;            // += c*KC*D at use
        sdst[j] = (unsigned)(size_t)&lx[0][z * (KC * D) + 4 * q];
    }
    const unsigned bufstride = (unsigned)(ZT * KC * D * sizeof(float));

    // accumulators: D row-subtiles x 2 N-halves, each 16x16 f32 (8 VGPRs)
    v8f acc[D][2];
    const v8f vzero = {0.f, 0.f, 0.f, 0.f, 0.f, 0.f, 0.f, 0.f};
#pragma unroll
    for (int s = 0; s < D; ++s) { acc[s][0] = vzero; acc[s][1] = vzero; }

    // Per-lane LDS base for A fragments: row m = s*16 + l16 -> (zLocal, i)
    int abase[D];
#pragma unroll
    for (int s = 0; s < D; ++s) {
        int r  = s * 16 + l16;
        int zL = r / D;
        int i  = r % D;
        abase[s] = zL * (KC * D) + i;
    }

    // Prologue: async-stage chunk 0 into buffer 0
#pragma unroll
    for (int j = 0; j < D; ++j)
        ASYNC_CP_B128(sdst[j], ssrc[j]);

#pragma unroll
    for (int c = 0; c < NCHUNK; ++c) {
        // Prefetch next chunk into the other buffer, then wait only for ours.
        if (c + 1 < NCHUNK) {
#pragma unroll
            for (int j = 0; j < D; ++j)
                ASYNC_CP_B128(sdst[j] + ((c + 1) & 1) * bufstride,
                              ssrc[j] + (size_t)(c + 1) * KC * D);
            WAIT_ASYNC(D);   // async loads complete in order: ours are done
        } else {
            WAIT_ASYNC(0);
        }
        __syncthreads();

        const float* __restrict__ lbuf = &lx[c & 1][0];
        const int kc = c * KC;

        // ---- K-chunk compute: 16 WMMA K-steps of 4 ----
#pragma unroll 4
        for (int kl = 0; kl < KC; kl += 4) {
            const int kA = kl + 2 * hi;           // local u of A.x / B.x
            const int kB = kc + kA;               // global u row of W

            // B fragments (4x16 K x N) for both N halves; W is L2-resident
            const float* wp = wblk + (size_t)kB * NCOL + wave * 16 + l16;
            v2f b0, b1;
            b0.x = wp[0];    b0.y = wp[NCOL];         // rows kB, kB+1
            b1.x = wp[128];  b1.y = wp[NCOL + 128];   // N-half 1

#pragma unroll
            for (int s = 0; s < D; ++s) {
                const int off = abase[s] + kA * D;
                v2f a;
                a.x = lbuf[off];        // X[m, kA]
                a.y = lbuf[off + D];    // X[m, kA+1]
                acc[s][0] = __builtin_amdgcn_wmma_f32_16x16x4_f32(
                    false, a, false, b0, (short)0, acc[s][0], false, false);
                acc[s][1] = __builtin_amdgcn_wmma_f32_16x16x4_f32(
                    false, a, false, b1, (short)0, acc[s][1], false, false);
            }
        }
        __syncthreads();   // protect buffer before next prefetch overwrites it
    }

    // ---- store: C[m,n] -> Y[z, n*D + i], scaled by 1/sqrt(256) ----
    const float scale = 0.0625f;
    float* __restrict__ yblk = y + (size_t)z0 * FEAT + xoff;
#pragma unroll
    for (int s = 0; s < D; ++s) {
#pragma unroll
        for (int nl = 0; nl < 2; ++nl) {
            const int ncol = nl * 128 + wave * 16 + l16;
#pragma unroll
            for (int v = 0; v < 8; ++v) {
                const int r  = s * 16 + v + 8 * hi;   // C row: M = vgpr + 8*hi
                const int zL = r / D;
                const int i  = r % D;
                yblk[(size_t)zL * FEAT + ncol * D + i] = acc[s][nl][v] * scale;
            }
        }
    }
}

extern "C" void kernel_launch(void* const* d_in, const int* in_sizes, int n_in,
                              void* d_out, int out_size, void* d_ws, size_t ws_size,
                              hipStream_t stream) {
    (void)in_sizes; (void)n_in; (void)d_ws; (void)ws_size; (void)out_size;
    const float* feats  = (const float*)d_in[0];
    const float* weight = (const float*)d_in[1];
    float* out = (float*)d_out;

    dim3 grid(BATCH / ZT);   // 3125, exact (50000 % 16 == 0)
    dim3 block(256);

    // xoff in floats: 0, 256*1, 256*1+256*3 = 1024 ; woff: 0, 65536, 131072
    linear_block_kernel<1><<<grid, block, 0, stream>>>(feats, weight, out,    0,      0);
    linear_block_kernel<3><<<grid, block, 0, stream>>>(feats, weight, out,  256,  65536);
    linear_block_kernel<5><<<grid, block, 0, stream>>>(feats, weight, out, 1024, 131072);
}